// AttModel_66700842107243
// MI455X (gfx1250) — compile-verified
//
#include <hip/hip_runtime.h>
#include <math.h>

typedef float v2f __attribute__((ext_vector_type(2)));
typedef float v8f __attribute__((ext_vector_type(8)));

#define PI_F 3.14159265358979323846f

// ---------------- workspace layout (in floats) ----------------
#define OFF_DCTKQ   0L            // 100  (10x10 DCT)
#define OFF_DCTM    256L          // 1225 (35x35 DCT)
#define OFF_SQ      2048L         // 256*240      src_query
#define OFF_SK      63488L        // 256*16*240   src_key
#define OFF_SV      1046528L      // 256*16*960   src_value
#define OFF_Q       4978688L      // 256*4*512    q projections
#define OFF_DATT    5502976L      // 256*4*960    attention output
#define OFF_X0      6486016L      // 256*48*40    gcn input
#define OFF_DOUT    6977536L      // 256*48*40    dct_out
#define OFF_T       7469056L      // 256*48*512   gemm temp (also aliases kout 256*4*16*512)
#define OFF_Y       13760512L     // 256*48*512   y buffer
#define OFF_H       20051968L     // 256*48*512   h buffer
#define OFF_K       OFF_T         // kout alias: dead before T/Y first written

// ---------------- DCT matrix construction ----------------
// Orthonormal DCT-II: dct[k,i] = w(k)*cos(pi*(i+0.5)*k/N); inverse == transpose.
__global__ void build_dct(float* __restrict__ dct_kq, float* __restrict__ dct_m35) {
    int t = blockIdx.x * blockDim.x + threadIdx.x;
    if (t < 100) {
        int k = t / 10, i = t % 10;
        float w = (k == 0) ? sqrtf(1.0f / 10.0f) : sqrtf(2.0f / 10.0f);
        dct_kq[t] = w * cosf(PI_F * (i + 0.5f) * (float)k / 10.0f);
    }
    int u = t - 128;
    if (u >= 0 && u < 1225) {
        int k = u / 35, i = u % 35;
        float w = (k == 0) ? sqrtf(1.0f / 35.0f) : sqrtf(2.0f / 35.0f);
        dct_m35[u] = w * cosf(PI_F * (i + 0.5f) * (float)k / 35.0f);
    }
}

// ---------------- attention preprocessing ----------------
// src: (256,50,48). Produces src_query (b,240), src_key (b,16,240), src_value (b,16,960)
__global__ void prep_attention(const float* __restrict__ src,
                               const float* __restrict__ dct_kq,
                               const float* __restrict__ dct_m35,
                               float* __restrict__ src_query,
                               float* __restrict__ src_key,
                               float* __restrict__ src_value) {
    __shared__ float s[50 * 48];
    int b = blockIdx.x;
    for (int i = threadIdx.x; i < 50 * 48; i += blockDim.x)
        s[i] = src[(long)b * 2400 + i];
    __syncthreads();
    // src_query[k*48+n] = sum_t dct_kq[k,t]*src[b,40+t,n]
    for (int e = threadIdx.x; e < 240; e += blockDim.x) {
        int k = e / 48, n = e % 48;
        float a = 0.f;
        for (int t = 0; t < 10; t++) a += dct_kq[k * 10 + t] * s[(40 + t) * 48 + n];
        src_query[(long)b * 240 + e] = a;
    }
    // src_key[v, k*48+n] = sum_t dct_kq[k,t]*src[b,v+t,n]
    for (int e = threadIdx.x; e < 16 * 240; e += blockDim.x) {
        int v = e / 240, r = e % 240, k = r / 48, n = r % 48;
        float a = 0.f;
        for (int t = 0; t < 10; t++) a += dct_kq[k * 10 + t] * s[(v + t) * 48 + n];
        src_key[(long)b * 3840 + e] = a;
    }
    // src_value[v, n*20+j] = sum_{t<35} dct_m35[j,t]*src[b,v+t,n]
    for (int e = threadIdx.x; e < 16 * 960; e += blockDim.x) {
        int v = e / 960, r = e % 960, n = r / 20, j = r % 20;
        float a = 0.f;
        for (int t = 0; t < 35; t++) a += dct_m35[j * 35 + t] * s[(v + t) * 48 + n];
        src_value[(long)b * 15360 + e] = a;
    }
}

// ---------------- WMMA f32 GEMM (N fixed at 512) ----------------
// Out[b,h] (m_rows x 512) = X[b] (m_rows x K) @ W[h] (K x 512) (+ bias[h] per column)
// grid: (512/(64*NTILES), B, H); block: 128 threads (4 waves).
// Each wave owns NTILES 16-col strips and MTILES 16-row strips.
// V_WMMA_F32_16X16X4_F32 layouts:
//   A 16x4:  lane L (L<16): M=L, vgpr{0,1}=K{0,1};  lane L>=16: M=L-16, K{2,3}
//   B 4x16:  lane L (L<16): N=L, vgpr{0,1}=K{0,1};  lane L>=16: N=L-16, K{2,3}
//   C 16x16: vgpr r: lanes0-15 -> M=r, lanes16-31 -> M=r+8; N=lane&15
// MASKED: rows >= m_rows are clamped to row 0 and zeroed via a 0/1 multiplier
// (branch-free; keeps EXEC all-ones through the K-loop as WMMA requires).
template <int MTILES, int NTILES, bool MASKED>
__global__ void __launch_bounds__(128)
gemm_wmma_f32(const float* __restrict__ X, long x_bstride, int m_rows,
              const float* __restrict__ W, long w_hstride,
              const float* __restrict__ bias, long bias_hstride,
              float* __restrict__ Out, long out_bstride, long out_hstride,
              int K) {
    constexpr int N = 512;
    const int cb = blockIdx.x;
    const int b = blockIdx.y;
    const int h = blockIdx.z;
    const int wave = threadIdx.x >> 5;
    const int lane = threadIdx.x & 31;
    const int halfsel = lane >> 4;          // 0: lanes 0-15, 1: lanes 16-31
    const int col0 = cb * (64 * NTILES) + wave * (16 * NTILES) + (lane & 15);

    const float* Xb = X + (long)b * x_bstride;
    const float* Wh = W + (long)h * w_hstride;

    // A-fragment row pointers (clamped when masked) + 0/1 masks
    const float* ap[MTILES];
    float amask[MTILES];
    for (int m = 0; m < MTILES; m++) {
        const int row = m * 16 + (lane & 15);
        int r = row;
        float msk = 1.0f;
        if (MASKED) {
            if (row >= m_rows) { r = 0; msk = 0.0f; }
        }
        ap[m] = Xb + (long)r * K + 2 * halfsel;
        amask[m] = msk;
    }
    // B-fragment column pointers
    const float* bp[NTILES];
    for (int n = 0; n < NTILES; n++)
        bp[n] = Wh + (long)(2 * halfsel) * N + col0 + n * 16;

    v8f acc[MTILES][NTILES];
    for (int m = 0; m < MTILES; m++)
        for (int n = 0; n < NTILES; n++)
            for (int i = 0; i < 8; i++) acc[m][n][i] = 0.0f;

    for (int k = 0; k < K; k += 4) {
        v2f bf[NTILES];
        for (int n = 0; n < NTILES; n++) {
            bf[n].x = bp[n][0];
            bf[n].y = bp[n][N];
            bp[n] += 4 * N;
        }
        v2f af[MTILES];
        for (int m = 0; m < MTILES; m++) {
            if (MASKED) {
                af[m].x = amask[m] * ap[m][0];
                af[m].y = amask[m] * ap[m][1];
            } else {
                af[m].x = ap[m][0];
                af[m].y = ap[m][1];
            }
            ap[m] += 4;
        }
        for (int m = 0; m < MTILES; m++)
            for (int n = 0; n < NTILES; n++)
                acc[m][n] = __builtin_amdgcn_wmma_f32_16x16x4_f32(
                    false, af[m], false, bf[n], (short)0, acc[m][n], false, false);
    }

    for (int n = 0; n < NTILES; n++) {
        const int col = col0 + n * 16;
        const float bv = bias ? bias[(long)h * bias_hstride + col] : 0.0f;
        float* Ob = Out + (long)b * out_bstride + (long)h * out_hstride + col;
        for (int m = 0; m < MTILES; m++) {
            for (int r = 0; r < 8; r++) {
                const int row = m * 16 + r + 8 * halfsel;
                if (!MASKED || row < m_rows) Ob[(long)row * N] = acc[m][n][r] + bv;
            }
        }
    }
}

// ---------------- attention score/normalize/value-mix ----------------
// block per (b,h): score[v]=q.k[v]+1e-15; att=score/sum; datt[j]=sum_v att[v]*src_value[b,v,j]
__global__ void attention(const float* __restrict__ qout,
                          const float* __restrict__ kout,
                          const float* __restrict__ src_value,
                          float* __restrict__ datt) {
    int bh = blockIdx.x;
    int b = bh >> 2;
    __shared__ float att[16];
    const float* q = qout + (long)bh * 512;
    if (threadIdx.x < 16) {
        const float* kv = kout + ((long)bh * 16 + threadIdx.x) * 512;
        float a = 0.f;
        for (int d = 0; d < 512; d++) a += q[d] * kv[d];
        att[threadIdx.x] = a + 1e-15f;
    }
    __syncthreads();
    if (threadIdx.x == 0) {
        float sum = 0.f;
        for (int v = 0; v < 16; v++) sum += att[v];
        float inv = 1.0f / sum;
        for (int v = 0; v < 16; v++) att[v] *= inv;
    }
    __syncthreads();
    for (int j = threadIdx.x; j < 960; j += blockDim.x) {
        float a = 0.f;
        const float* sv = src_value + (long)b * 15360 + j;
        for (int v = 0; v < 16; v++) a += att[v] * sv[v * 960];
        datt[(long)bh * 960 + j] = a;
    }
}

// ---------------- build GCN input x0 = concat(dct_in, dct_att@Ww0+bw0) ----------------
__global__ void build_x0(const float* __restrict__ src,
                         const float* __restrict__ dct_m35,
                         const float* __restrict__ datt,
                         const float* __restrict__ Ww0,
                         const float* __restrict__ bw0,
                         float* __restrict__ x0) {
    long t = (long)blockIdx.x * blockDim.x + threadIdx.x;
    if (t >= 256L * 48 * 40) return;
    int c = (int)(t % 40);
    int n = (int)((t / 40) % 48);
    int b = (int)(t / (40 * 48));
    float a;
    if (c < 20) {
        // dct_in[b,n,j] = sum_t dct_m35[j,t]*src[b,pad_idx[t],n]; pad_idx = 40..49, then 49x25
        a = 0.f;
        for (int tt = 0; tt < 35; tt++) {
            int row = (tt < 10) ? (40 + tt) : 49;
            a += dct_m35[c * 35 + tt] * src[((long)b * 50 + row) * 48 + n];
        }
    } else {
        int cc = c - 20;
        a = bw0[cc];
        // in80[b,n,o=h*20+j] = datt[b,h,n*20+j]
        for (int o = 0; o < 80; o++) {
            int hh = o / 20, j = o % 20;
            a += datt[((long)b * 4 + hh) * 960 + n * 20 + j] * Ww0[o * 20 + cc];
        }
    }
    x0[t] = a;
}

// ---------------- A-mix + BN + tanh (+optional residual) ----------------
// block per (b,n), 512 threads (f). out[b,n,f] = tanh((sum_m A[n,m]T[b,m,f] + bias[f])*g*inv_std + bt) [+ res]
__global__ void mix_bn_tanh(const float* __restrict__ T,
                            const float* __restrict__ A,
                            const float* __restrict__ bias,
                            const float* __restrict__ g,
                            const float* __restrict__ bt,
                            const float* __restrict__ res,
                            float* __restrict__ Out) {
    int bn = blockIdx.x;
    int b = bn / 48, n = bn % 48;
    int f = threadIdx.x;
    const float* Tb = T + (long)b * 48 * 512;
    float acc = 0.f;
    for (int m = 0; m < 48; m++) acc += A[n * 48 + m] * Tb[m * 512 + f];
    const float inv_std = 0.9999950000374997f; // 1/sqrt(1+1e-5)
    float v = (acc + bias[f]) * (g[n * 512 + f] * inv_std) + bt[n * 512 + f];
    v = tanhf(v);
    long idx = (long)b * 48 * 512 + n * 512 + f;
    if (res) v += res[idx];
    Out[idx] = v;
}

// ---------------- gc7: Y(48x512)@gc7_W(512x40) ----------------
__global__ void gc7_xw(const float* __restrict__ Y, const float* __restrict__ W,
                       float* __restrict__ G) {
    long t = (long)blockIdx.x * blockDim.x + threadIdx.x;
    if (t >= 256L * 48 * 40) return;
    int c = (int)(t % 40);
    long bn = t / 40;
    const float* y = Y + bn * 512;
    float a = 0.f;
    for (int f = 0; f < 512; f++) a += y[f] * W[f * 40 + c];
    G[t] = a;
}

// dct_out[b,n,c] = sum_m gc7_A[n,m]*G[b,m,c] + gc7_b[c] + x0[b,n,c]
__global__ void mix7(const float* __restrict__ G, const float* __restrict__ A,
                     const float* __restrict__ bias, const float* __restrict__ x0,
                     float* __restrict__ dct_out) {
    long t = (long)blockIdx.x * blockDim.x + threadIdx.x;
    if (t >= 256L * 48 * 40) return;
    int c = (int)(t % 40);
    int n = (int)((t / 40) % 48);
    long b = t / (48 * 40);
    float a = bias[c] + x0[t];
    const float* Gb = G + b * 48 * 40;
    for (int m = 0; m < 48; m++) a += A[n * 48 + m] * Gb[m * 40 + c];
    dct_out[t] = a;
}

// ---------------- final iDCT: out[b,t,n] = sum_{j<20} dct_m35[j,t]*dct_out[b,n,j] ----------------
__global__ void idct_out(const float* __restrict__ dct_out,
                         const float* __restrict__ dct_m35,
                         float* __restrict__ out) {
    long t = (long)blockIdx.x * blockDim.x + threadIdx.x;
    if (t >= 256L * 35 * 48) return;
    int n = (int)(t % 48);
    int tt = (int)((t / 48) % 35);
    long b = t / (35 * 48);
    const float* d = dct_out + (b * 48 + n) * 40;
    float a = 0.f;
    for (int j = 0; j < 20; j++) a += dct_m35[j * 35 + tt] * d[j];
    out[t] = a;
}

extern "C" void kernel_launch(void* const* d_in, const int* in_sizes, int n_in,
                              void* d_out, int out_size, void* d_ws, size_t ws_size,
                              hipStream_t stream) {
    const float* src  = (const float*)d_in[0];
    // d_in[1..4] = dct_n, output_n, input_n, itera (constants; baked in)
    const float* Wq   = (const float*)d_in[5];
    const float* bq   = (const float*)d_in[6];
    const float* Wk   = (const float*)d_in[7];
    const float* bk   = (const float*)d_in[8];
    const float* Ww0  = (const float*)d_in[9];
    const float* bw0  = (const float*)d_in[10];
    const float* gc1W = (const float*)d_in[11];
    const float* gc1A = (const float*)d_in[12];
    const float* gc1b = (const float*)d_in[13];
    const float* bn1g = (const float*)d_in[14];
    const float* bn1b = (const float*)d_in[15];
    const float* blkW = (const float*)d_in[16];
    const float* blkA = (const float*)d_in[17];
    const float* blkb = (const float*)d_in[18];
    const float* blkg = (const float*)d_in[19];
    const float* blkbt= (const float*)d_in[20];
    const float* gc7W = (const float*)d_in[21];
    const float* gc7A = (const float*)d_in[22];
    const float* gc7b = (const float*)d_in[23];

    float* ws  = (float*)d_ws;
    float* out = (float*)d_out;

    float* dctkq = ws + OFF_DCTKQ;
    float* dctm  = ws + OFF_DCTM;
    float* sq    = ws + OFF_SQ;
    float* sk    = ws + OFF_SK;
    float* sv    = ws + OFF_SV;
    float* qo    = ws + OFF_Q;
    float* ko    = ws + OFF_K;     // aliases T region (dead before T written)
    float* datt  = ws + OFF_DATT;
    float* x0    = ws + OFF_X0;
    float* dcto  = ws + OFF_DOUT;
    float* T     = ws + OFF_T;
    float* Y     = ws + OFF_Y;
    float* H     = ws + OFF_H;

    // 1. DCT matrices
    build_dct<<<6, 256, 0, stream>>>(dctkq, dctm);
    // 2. query/key/value prep
    prep_attention<<<256, 256, 0, stream>>>(src, dctkq, dctm, sq, sk, sv);
    // 3. q projection: (1x240)@(240x512) per (b,h)  [WMMA, masked to 1 row]
    gemm_wmma_f32<1, 2, true><<<dim3(4, 256, 4), 128, 0, stream>>>(
        sq, 240, 1, Wq, 240L * 512, bq, 512, qo, 4L * 512, 512, 240);
    // 4. k projection: (16x240)@(240x512) per (b,h)  [exact 16-row tile, unmasked]
    gemm_wmma_f32<1, 2, false><<<dim3(4, 256, 4), 128, 0, stream>>>(
        sk, 3840, 16, Wk, 240L * 512, bk, 512, ko, 4L * 16 * 512, 16L * 512, 240);
    // 5. attention
    attention<<<1024, 256, 0, stream>>>(qo, ko, sv, datt);
    // 6. GCN input
    {
        long n = 256L * 48 * 40;
        build_x0<<<(unsigned)((n + 255) / 256), 256, 0, stream>>>(src, dctm, datt, Ww0, bw0, x0);
    }
    // 7. gc1: (48x40)@(40x512)  [48 rows = exact 3 M-tiles, unmasked]
    gemm_wmma_f32<3, 2, false><<<dim3(4, 256, 1), 128, 0, stream>>>(
        x0, 48L * 40, 48, gc1W, 0, nullptr, 0, T, 48L * 512, 0, 40);
    mix_bn_tanh<<<256 * 48, 512, 0, stream>>>(T, gc1A, gc1b, bn1g, bn1b, nullptr, Y);
    // 8. two stages x two layers, residual after each stage
    for (int st = 0; st < 2; st++) {
        const float* W0 = blkW + (long)(st * 2 + 0) * 512 * 512;
        const float* W1 = blkW + (long)(st * 2 + 1) * 512 * 512;
        gemm_wmma_f32<3, 2, false><<<dim3(4, 256, 1), 128, 0, stream>>>(
            Y, 48L * 512, 48, W0, 0, nullptr, 0, T, 48L * 512, 0, 512);
        mix_bn_tanh<<<256 * 48, 512, 0, stream>>>(
            T, blkA + (st * 2 + 0) * 2304, blkb + (st * 2 + 0) * 512,
            blkg + (long)(st * 2 + 0) * 24576, blkbt + (long)(st * 2 + 0) * 24576, nullptr, H);
        gemm_wmma_f32<3, 2, false><<<dim3(4, 256, 1), 128, 0, stream>>>(
            H, 48L * 512, 48, W1, 0, nullptr, 0, T, 48L * 512, 0, 512);
        mix_bn_tanh<<<256 * 48, 512, 0, stream>>>(
            T, blkA + (st * 2 + 1) * 2304, blkb + (st * 2 + 1) * 512,
            blkg + (long)(st * 2 + 1) * 24576, blkbt + (long)(st * 2 + 1) * 24576, Y, Y);
    }
    // 9. gc7 + residual dct_in, then iDCT to output
    {
        long n = 256L * 48 * 40;
        gc7_xw<<<(unsigned)((n + 255) / 256), 256, 0, stream>>>(Y, gc7W, T);
        mix7<<<(unsigned)((n + 255) / 256), 256, 0, stream>>>(T, gc7A, gc7b, x0, dcto);
        long no = 256L * 35 * 48;
        idct_out<<<(unsigned)((no + 255) / 256), 256, 0, stream>>>(dcto, dctm, out);
    }
}